// UnifiedDepthModel_38809324487106
// MI455X (gfx1250) — compile-verified
//
#include <hip/hip_runtime.h>
#include <hip/hip_bf16.h>

// ---------------------------------------------------------------------------
// Stereo depth model (cost volume + window attention CRF + conv head)
// fp32 end-to-end; all GEMMs + conv3x3 (implicit GEMM) on
// V_WMMA_F32_16X16X4_F32 (wave32). GEMM A-tiles staged via the Tensor Data
// Mover (TENSOR_LOAD_TO_LDS) with LDS pitch padding done by the TDM itself.
// Workspace requirement: ~472 MB (450 MiB) (layout in kernel_launch).
// ---------------------------------------------------------------------------

#define BB    4
#define CC    128
#define HH    128
#define WW2   128
#define HWN   (HH * WW2)        // 16384
#define NTOK  (BB * HWN)        // 65536
#define DD    48
#define NHD   8
#define HDIM  16
#define HIDC  512

typedef __attribute__((ext_vector_type(2))) float v2f;
typedef __attribute__((ext_vector_type(8))) float v8f;

__device__ inline v8f wmma_f32(v2f a, v2f b, v8f c) {
  // (neg_a, A, neg_b, B, c_mod, C, reuse_a, reuse_b)
  return __builtin_amdgcn_wmma_f32_16x16x4_f32(false, a, false, b, (short)0, c,
                                               false, false);
}

__device__ inline float gelu_f(float x) {
  return 0.5f * x * (1.f + erff(x * 0.70710678118654752f));
}

// ------------------------- TDM tile staging (CDNA5) ------------------------
#if __has_builtin(__builtin_amdgcn_tensor_load_to_lds) && \
    __has_builtin(__builtin_amdgcn_s_wait_tensorcnt)
#define HAVE_TDM 1
typedef __attribute__((ext_vector_type(4))) unsigned int u32x4;
typedef __attribute__((ext_vector_type(8))) int i32x8;
typedef __attribute__((ext_vector_type(4))) int i32x4;

// DMA a 64x64 f32 tile (row stride = strideElems) into LDS at byte offset
// lds_off, inserting 1 DWORD of padding every 64 DWORDs -> pitch-65 layout.
__device__ inline void tdm_load_64x64(const float* gsrc, unsigned lds_off,
                                      int strideElems) {
  unsigned long long ga = (unsigned long long)(uintptr_t)gsrc;
  u32x4 g0;
  g0[0] = 1u;                                    // count=1 (valid user D#)
  g0[1] = lds_off;                               // LDS byte address
  g0[2] = (unsigned)(ga & 0xffffffffu);          // global_addr[31:0]
  g0[3] = (unsigned)((ga >> 32) & 0x1ffffffu)    // global_addr[56:32]
          | (2u << 30);                          // type=2 ("image")
  const unsigned td0 = 1u << 30, td1 = 1u << 30; // huge dims: no OOB clipping
  i32x8 g1;
  g1[0] = (2 << 16)                              // data_size = 4 bytes
          | (1 << 20)                            // pad_enable
          | (5 << 22)                            // pad_interval = 64 DWORDs
          | (0 << 25);                           // pad_amount   = 1 DWORD
  g1[1] = (int)((td0 & 0xffffu) << 16);          // tensor_dim0[15:0]
  g1[2] = (int)(((td0 >> 16) & 0xffffu) | ((td1 & 0xffffu) << 16));
  g1[3] = (int)(((td1 >> 16) & 0xffffu) | (64u << 16));  // tile_dim0 = 64
  g1[4] = 64;                                    // tile_dim1=64, tile_dim2=0
  g1[5] = strideElems;                           // tensor_dim0_stride lo32
  g1[6] = 0;
  g1[7] = 0;
  i32x4 z4 = {0, 0, 0, 0};
  i32x8 z8 = {0, 0, 0, 0, 0, 0, 0, 0};
  // 6-arg variant (clang expects 6): (g0, g1, g2, g3, g4, cpol)
  __builtin_amdgcn_tensor_load_to_lds(g0, g1, z4, z4, z8, 0);
}
#else
#define HAVE_TDM 0
#endif

// --------------------------- layout shuffles -------------------------------

// feat_l [B,C,H,W] -> tokens [B*HW, C]
__global__ void k_tokens(const float* __restrict__ feat, float* __restrict__ tok) {
  int idx = blockIdx.x * blockDim.x + threadIdx.x;     // over NTOK*CC
  int c = idx & (CC - 1);
  int t = idx >> 7;
  int b = t >> 14;
  int hw = t & (HWN - 1);
  tok[idx] = feat[((size_t)(b * CC + c)) * HWN + hw];
}

// x [B*HW, C] -> crf [B,C,H,W]
__global__ void k_tok2nchw(const float* __restrict__ x, float* __restrict__ out) {
  int idx = blockIdx.x * blockDim.x + threadIdx.x;     // over B*C*HW
  int hw = idx & (HWN - 1);
  int bc = idx >> 14;
  int c = bc & (CC - 1);
  int b = bc >> 7;
  out[idx] = x[((size_t)b * HWN + hw) * CC + c];
}

// ------------------------------ cost volume --------------------------------
__global__ void k_cost(const float* __restrict__ fl, const float* __restrict__ fr,
                       float* __restrict__ cost) {
  __shared__ float sl[WW2];
  __shared__ float sr[WW2];
  int bh = blockIdx.x;            // B*H blocks
  int b = bh >> 7;
  int h = bh & 127;
  int w = threadIdx.x;            // 128 threads
  float acc[DD];
#pragma unroll
  for (int d = 0; d < DD; d++) acc[d] = 0.f;
  for (int c = 0; c < CC; c++) {
    __syncthreads();
    sl[w] = fl[((size_t)(b * CC + c)) * HWN + h * WW2 + w];
    sr[w] = fr[((size_t)(b * CC + c)) * HWN + h * WW2 + w];
    __syncthreads();
    float a = sl[w];
#pragma unroll
    for (int d = 0; d < DD; d++) {
      float bv = sr[(w >= d) ? (w - d) : 0];
      acc[d] += (w >= d) ? a * bv : 0.f;
    }
  }
#pragma unroll
  for (int d = 0; d < DD; d++)
    cost[((size_t)(b * DD + d)) * HWN + h * WW2 + w] = acc[d] * (1.f / 128.f);
}

// ------------------------------- layernorm ---------------------------------
__global__ void k_ln(const float* __restrict__ in, const float* __restrict__ g,
                     const float* __restrict__ bt, float* __restrict__ out, int part) {
  __shared__ float red[128];
  int t = blockIdx.x;
  int c = threadIdx.x;
  float v = in[(size_t)t * CC + c];
  red[c] = v;
  __syncthreads();
#pragma unroll
  for (int s = 64; s > 0; s >>= 1) {
    if (c < s) red[c] += red[c + s];
    __syncthreads();
  }
  float mean = red[0] * (1.f / 128.f);
  __syncthreads();
  float d = v - mean;
  red[c] = d * d;
  __syncthreads();
#pragma unroll
  for (int s = 64; s > 0; s >>= 1) {
    if (c < s) red[c] += red[c + s];
    __syncthreads();
  }
  float var = red[0] * (1.f / 128.f);
  float y = d * rsqrtf(var + 1e-5f) * g[c] + bt[c];
  size_t orow = (size_t)t;
  if (part) {
    int b = t >> 14, hw = t & (HWN - 1);
    int hh = hw >> 7, wv = hw & 127;
    int win = b * 256 + (hh >> 3) * 16 + (wv >> 3);
    int inner = (hh & 7) * 8 + (wv & 7);
    orow = (size_t)win * 64 + inner;
  }
  out[orow * CC + c] = y;
}

// ------------------------------ generic GEMM -------------------------------
// out = act(A[M,K] @ W[K,N] + bias) [+ res];  block = 4 waves -> 64x64 tile.
// A tile staged via TDM (pitch-65 padding by the DMA engine); B tile staged
// cooperatively with float4 loads. Inner loop: ds_load + wmma only.
__global__ void k_gemm(const float* __restrict__ A, const float* __restrict__ Wm,
                       const float* __restrict__ bias, const float* __restrict__ res,
                       float* __restrict__ out, int M, int K, int N, int act,
                       int remap) {
  (void)M;
  __shared__ float As[64][65];
  __shared__ float Bs[64][65];
  int tidx = threadIdx.x;
  int wave = tidx >> 5;
  int lane = tidx & 31;
  int tileM = blockIdx.x;
  int tileN = blockIdx.y;
  int row0 = tileM * 64 + wave * 16;
  int col0 = tileN * 64;
  int l15 = lane & 15;
  int khalf = (lane >> 4) << 1;          // 0 or 2
  v8f acc[4] = {};
  for (int kc = 0; kc < K; kc += 64) {
    __syncthreads();
#if HAVE_TDM
    if (tidx < 32) {
      tdm_load_64x64(A + ((size_t)tileM * 64) * K + kc,
                     (unsigned)(uintptr_t)&As[0][0], K);
      __builtin_amdgcn_s_wait_tensorcnt(0);
    }
#else
    for (int i = tidx; i < 64 * 16; i += 128) {
      int r = i >> 4, c4 = (i & 15) << 2;
      const float4 vA =
          *(const float4*)(A + ((size_t)(tileM * 64 + r)) * K + kc + c4);
      As[r][c4 + 0] = vA.x; As[r][c4 + 1] = vA.y;
      As[r][c4 + 2] = vA.z; As[r][c4 + 3] = vA.w;
    }
#endif
    for (int i = tidx; i < 64 * 16; i += 128) {
      int r = i >> 4, c4 = (i & 15) << 2;
      const float4 vB = *(const float4*)(Wm + ((size_t)(kc + r)) * N + col0 + c4);
      Bs[r][c4 + 0] = vB.x; Bs[r][c4 + 1] = vB.y;
      Bs[r][c4 + 2] = vB.z; Bs[r][c4 + 3] = vB.w;
    }
    __syncthreads();
#pragma unroll 4
    for (int k4 = 0; k4 < 64; k4 += 4) {
      v2f a;
      a.x = As[wave * 16 + l15][k4 + khalf];
      a.y = As[wave * 16 + l15][k4 + khalf + 1];
#pragma unroll
      for (int nj = 0; nj < 4; nj++) {
        v2f bf;
        bf.x = Bs[k4 + khalf][nj * 16 + l15];
        bf.y = Bs[k4 + khalf + 1][nj * 16 + l15];
        acc[nj] = wmma_f32(a, bf, acc[nj]);
      }
    }
  }
  int chalf = (lane >> 4) << 3;
#pragma unroll
  for (int nj = 0; nj < 4; nj++) {
    int ncol = col0 + nj * 16 + l15;
    float bv = bias ? bias[ncol] : 0.f;
#pragma unroll
    for (int i = 0; i < 8; i++) {
      int r = row0 + i + chalf;
      size_t orow = (size_t)r;
      if (remap) {
        int win = r >> 6, inner = r & 63;
        int b = win >> 8, wi = win & 255;
        int hh = (wi >> 4) * 8 + (inner >> 3);
        int wv = (wi & 15) * 8 + (inner & 7);
        orow = (size_t)b * HWN + hh * WW2 + wv;
      }
      float o = acc[nj][i] + bv;
      if (act == 1) o = gelu_f(o);
      if (res) o += res[orow * N + ncol];
      out[orow * N + ncol] = o;
    }
  }
}

// ------------------------------- attention ---------------------------------
// one wave per (window, head): S = q k^T * 0.25 + rpb, softmax, O = P v
__global__ void k_attn(const float* __restrict__ qb, const float* __restrict__ kb,
                       const float* __restrict__ vb, const float* __restrict__ rpb,
                       float* __restrict__ ob) {
  __shared__ float P[64][65];
  int blk = blockIdx.x;            // Bw*NH = 8192
  int win = blk >> 3;
  int h = blk & 7;
  int lane = threadIdx.x & 31;
  int l15 = lane & 15;
  int khalf = (lane >> 4) << 1;
  int chalf = (lane >> 4) << 3;
  const size_t base = (size_t)win * 64 * CC + (size_t)h * HDIM;

  v8f S[4][4] = {};
#pragma unroll
  for (int kk = 0; kk < HDIM; kk += 4) {
    v2f bf[4];
#pragma unroll
    for (int nj = 0; nj < 4; nj++) {
      bf[nj].x = kb[base + ((size_t)(nj * 16 + l15)) * CC + kk + khalf];
      bf[nj].y = kb[base + ((size_t)(nj * 16 + l15)) * CC + kk + khalf + 1];
    }
#pragma unroll
    for (int mi = 0; mi < 4; mi++) {
      v2f a;
      a.x = qb[base + ((size_t)(mi * 16 + l15)) * CC + kk + khalf];
      a.y = qb[base + ((size_t)(mi * 16 + l15)) * CC + kk + khalf + 1];
#pragma unroll
      for (int nj = 0; nj < 4; nj++) S[mi][nj] = wmma_f32(a, bf[nj], S[mi][nj]);
    }
  }

#pragma unroll
  for (int mi = 0; mi < 4; mi++) {
#pragma unroll
    for (int i = 0; i < 8; i++) {
      int m = mi * 16 + i + chalf;               // row token in window
      float mx = -3.0e38f;
#pragma unroll
      for (int nj = 0; nj < 4; nj++) {
        int n = nj * 16 + l15;                   // col token in window
        int di = (m >> 3) - (n >> 3) + 7;
        int dj = (m & 7) - (n & 7) + 7;
        float bias = rpb[(di * 15 + dj) * NHD + h];
        float logit = S[mi][nj][i] * 0.25f + bias;
        S[mi][nj][i] = logit;
        mx = fmaxf(mx, logit);
      }
#pragma unroll
      for (int xm = 1; xm < 16; xm <<= 1) mx = fmaxf(mx, __shfl_xor(mx, xm, 32));
      float ssum = 0.f;
#pragma unroll
      for (int nj = 0; nj < 4; nj++) {
        float e = __expf(S[mi][nj][i] - mx);
        S[mi][nj][i] = e;
        ssum += e;
      }
#pragma unroll
      for (int xm = 1; xm < 16; xm <<= 1) ssum += __shfl_xor(ssum, xm, 32);
      float inv = 1.f / ssum;
#pragma unroll
      for (int nj = 0; nj < 4; nj++) P[m][nj * 16 + l15] = S[mi][nj][i] * inv;
    }
  }
  __syncthreads();

  v8f O[4] = {};
#pragma unroll
  for (int kk = 0; kk < 64; kk += 4) {
    v2f bf;
    bf.x = vb[base + ((size_t)(kk + khalf)) * CC + l15];
    bf.y = vb[base + ((size_t)(kk + khalf + 1)) * CC + l15];
#pragma unroll
    for (int mi = 0; mi < 4; mi++) {
      v2f a;
      a.x = P[mi * 16 + l15][kk + khalf];
      a.y = P[mi * 16 + l15][kk + khalf + 1];
      O[mi] = wmma_f32(a, bf, O[mi]);
    }
  }
#pragma unroll
  for (int mi = 0; mi < 4; mi++)
#pragma unroll
    for (int i = 0; i < 8; i++)
      ob[base + ((size_t)(mi * 16 + i + chalf)) * CC + l15] = O[mi][i];
}

// ------------------ conv3x3 + BN + ReLU as implicit GEMM -------------------
// M = 128 pixels of one image row, N = 96 (88 real oc, zero-padded),
// K = 176*9 chunked as 22 x (8 ic * 9 taps = 72). 8 waves: wave = M-tile.
#define ICB 8
__global__ void k_conv3(const float* __restrict__ crf, const float* __restrict__ costb,
                        const float* __restrict__ wgt, const float* __restrict__ b1,
                        const float* __restrict__ bng, const float* __restrict__ bnb,
                        float* __restrict__ h1) {
  __shared__ float sIn[ICB][3][WW2 + 2];   // 12.5 KB (zero-padded halo)
  __shared__ float sW[72][97];             // 27.9 KB re-ordered weight tile
  int bh = blockIdx.x;                     // B*H
  int h = bh & 127;
  int b = bh >> 7;
  int tid = threadIdx.x;                   // 256 threads = 8 waves
  int wave = tid >> 5;
  int lane = tid & 31;
  int l15 = lane & 15;
  int khalf = (lane >> 4) << 1;
  int chalf = (lane >> 4) << 3;
  int w0 = wave * 16;                      // pixel base of this wave's M-tile
  v8f acc[6] = {};
  for (int ic0 = 0; ic0 < 176; ic0 += ICB) {
    __syncthreads();
    // stage input: ICB channels x 3 rows x 130 cols (zero halo)
    for (int i = tid; i < ICB * 3 * (WW2 + 2); i += 256) {
      int col = i % (WW2 + 2);
      int rem = i / (WW2 + 2);
      int dy = rem % 3;
      int icq = rem / 3;
      int ic = ic0 + icq;
      int hy = h + dy - 1;
      int wx = col - 1;
      float vld = 0.f;
      if (hy >= 0 && hy < HH && wx >= 0 && wx < WW2) {
        const float* src = (ic < 128)
                               ? crf + ((size_t)(b * 128 + ic)) * HWN
                               : costb + ((size_t)(b * DD + ic - 128)) * HWN;
        vld = src[hy * WW2 + wx];
      }
      sIn[icq][dy][col] = vld;
    }
    // stage weights re-ordered to K-major: sW[icq*9+t9][oc]
    for (int i = tid; i < 72 * 96; i += 256) {
      int oc = i % 96;
      int kkq = i / 96;
      int icq = kkq / 9;
      int t9 = kkq - icq * 9;
      float wv = 0.f;
      if (oc < 88) wv = wgt[((size_t)oc * 176 + ic0 + icq) * 9 + t9];
      sW[kkq][oc] = wv;
    }
    __syncthreads();
    for (int k4 = 0; k4 < 72; k4 += 4) {
      int kk = k4 + khalf;
      int icA = kk / 9, r9a = kk - icA * 9;
      int dyA = r9a / 3, dxA = r9a - dyA * 3;
      int kk1 = kk + 1;
      int icB = kk1 / 9, r9b = kk1 - icB * 9;
      int dyB = r9b / 3, dxB = r9b - dyB * 3;
      v2f a;
      a.x = sIn[icA][dyA][w0 + l15 + dxA];
      a.y = sIn[icB][dyB][w0 + l15 + dxB];
#pragma unroll
      for (int nt = 0; nt < 6; nt++) {
        v2f bf;
        bf.x = sW[kk][nt * 16 + l15];
        bf.y = sW[kk1][nt * 16 + l15];
        acc[nt] = wmma_f32(a, bf, acc[nt]);
      }
    }
  }
  const float rs = rsqrtf(1.f + 1e-5f);
#pragma unroll
  for (int nt = 0; nt < 6; nt++) {
    int oc = nt * 16 + l15;
    if (oc < 88) {
      float gg = bng[oc], bb = bnb[oc], bi = b1[oc];
#pragma unroll
      for (int i = 0; i < 8; i++) {
        int pix = w0 + i + chalf;
        float r = (acc[nt][i] + bi) * rs * gg + bb;
        h1[((size_t)(b * 88 + oc)) * HWN + h * WW2 + pix] = fmaxf(r, 0.f);
      }
    }
  }
}

// -------------------- conv1x1 + softmax + disparity ------------------------
__global__ void k_head(const float* __restrict__ h1, const float* __restrict__ w2,
                       const float* __restrict__ b2, float* __restrict__ outp) {
  __shared__ float wsm[DD * 88];
  int id = blockIdx.x;                 // B*H
  int h = id & 127;
  int b = id >> 7;
  for (int i = threadIdx.x; i < DD * 88; i += 128) wsm[i] = w2[i];
  __syncthreads();
  int w = threadIdx.x;
  float pv[DD];
#pragma unroll
  for (int d = 0; d < DD; d++) pv[d] = b2[d];
  for (int ic = 0; ic < 88; ic++) {
    float x = h1[((size_t)(b * 88 + ic)) * HWN + h * WW2 + w];
#pragma unroll
    for (int d = 0; d < DD; d++) pv[d] += x * wsm[d * 88 + ic];
  }
  float mx = -3.0e38f;
#pragma unroll
  for (int d = 0; d < DD; d++) mx = fmaxf(mx, pv[d]);
  float s = 0.f, num = 0.f;
#pragma unroll
  for (int d = 0; d < DD; d++) {
    float e = __expf(pv[d] - mx);
    s += e;
    num += e * (float)d;
  }
  outp[(size_t)b * HWN + h * WW2 + w] = num / s * 4.f;
}

// ------------------------------ launcher -----------------------------------
extern "C" void kernel_launch(void* const* d_in, const int* in_sizes, int n_in,
                              void* d_out, int out_size, void* d_ws, size_t ws_size,
                              hipStream_t stream) {
  (void)in_sizes; (void)n_in; (void)out_size; (void)ws_size;
  const float* feat_l = (const float*)d_in[0];
  const float* feat_r = (const float*)d_in[1];
  const float* wq = (const float*)d_in[2];
  const float* bq = (const float*)d_in[3];
  const float* wk = (const float*)d_in[4];
  const float* bk = (const float*)d_in[5];
  const float* wv = (const float*)d_in[6];
  const float* bv = (const float*)d_in[7];
  const float* wp = (const float*)d_in[8];
  const float* bp = (const float*)d_in[9];
  const float* rpb = (const float*)d_in[10];
  const float* ln1_g = (const float*)d_in[11];
  const float* ln1_b = (const float*)d_in[12];
  const float* ln2_g = (const float*)d_in[13];
  const float* ln2_b = (const float*)d_in[14];
  const float* mlp_w1 = (const float*)d_in[15];
  const float* mlp_b1 = (const float*)d_in[16];
  const float* mlp_w2 = (const float*)d_in[17];
  const float* un_w1 = (const float*)d_in[19];
  const float* un_b1 = (const float*)d_in[20];
  const float* un_w2 = (const float*)d_in[21];
  const float* un_b2 = (const float*)d_in[22];
  const float* dp_w1 = (const float*)d_in[23];
  const float* dp_b1 = (const float*)d_in[24];
  const float* bn_g = (const float*)d_in[25];
  const float* bn_b = (const float*)d_in[26];
  const float* dp_w2 = (const float*)d_in[27];
  const float* dp_b2 = (const float*)d_in[28];
  float* outp = (float*)d_out;

  float* base = (float*)d_ws;
  const size_t NC = (size_t)NTOK * CC;               // 8.39M floats
  float* tokens = base;                              // [NTOK, C]
  float* nf     = base + 1 * NC;                     // LN1(win_part); later xn
  float* unary  = base + 2 * NC;
  float* tmp    = base + 3 * NC;                     // unary hidden; later attn out
  float* qb     = base + 4 * NC;
  float* kb     = base + 5 * NC;
  float* vbuf   = base + 6 * NC;
  float* xbuf   = base + 7 * NC;
  float* crf    = base + 8 * NC;                     // NCHW
  float* hid512 = base + 9 * NC;                     // [NTOK, 512] = 4*NC
  float* costb  = base + 13 * NC;                    // [B, D, H, W]
  float* h1b    = costb + (size_t)BB * DD * HWN;     // [B, 88, H, W]

  dim3 blk128(128);

  k_tokens<<<(NTOK * CC) / 256, 256, 0, stream>>>(feat_l, tokens);
  k_cost<<<BB * HH, blk128, 0, stream>>>(feat_l, feat_r, costb);
  k_gemm<<<dim3(NTOK / 64, CC / 64), blk128, 0, stream>>>(
      tokens, un_w1, un_b1, nullptr, tmp, NTOK, CC, CC, 1, 0);
  k_gemm<<<dim3(NTOK / 64, CC / 64), blk128, 0, stream>>>(
      tmp, un_w2, un_b2, nullptr, unary, NTOK, CC, CC, 0, 0);
  k_ln<<<NTOK, blk128, 0, stream>>>(tokens, ln1_g, ln1_b, nf, 1);
  k_gemm<<<dim3(NTOK / 64, CC / 64), blk128, 0, stream>>>(
      nf, wq, bq, nullptr, qb, NTOK, CC, CC, 0, 0);
  k_gemm<<<dim3(NTOK / 64, CC / 64), blk128, 0, stream>>>(
      nf, wk, bk, nullptr, kb, NTOK, CC, CC, 0, 0);
  k_gemm<<<dim3(NTOK / 64, CC / 64), blk128, 0, stream>>>(
      nf, wv, bv, nullptr, vbuf, NTOK, CC, CC, 0, 0);
  k_attn<<<1024 * NHD, 32, 0, stream>>>(qb, kb, vbuf, rpb, tmp);
  k_gemm<<<dim3(NTOK / 64, CC / 64), blk128, 0, stream>>>(
      tmp, wp, bp, unary, xbuf, NTOK, CC, CC, 0, 1);
  k_ln<<<NTOK, blk128, 0, stream>>>(xbuf, ln2_g, ln2_b, nf, 0);
  k_gemm<<<dim3(NTOK / 64, HIDC / 64), blk128, 0, stream>>>(
      nf, mlp_w1, mlp_b1, nullptr, hid512, NTOK, CC, HIDC, 1, 0);
  k_gemm<<<dim3(NTOK / 64, CC / 64), blk128, 0, stream>>>(
      hid512, mlp_w2, nullptr, xbuf, xbuf, NTOK, HIDC, CC, 0, 0);
  k_tok2nchw<<<(NTOK * CC) / 256, 256, 0, stream>>>(xbuf, crf);
  k_conv3<<<BB * HH, 256, 0, stream>>>(crf, costb, dp_w1, dp_b1, bn_g, bn_b, h1b);
  k_head<<<BB * HH, blk128, 0, stream>>>(h1b, dp_w2, dp_b2, outp);
}